// Attn_63230508532520
// MI455X (gfx1250) — compile-verified
//
#include <hip/hip_runtime.h>
#include <hip/hip_bf16.h>
#include <math.h>

// Problem sizes (fixed by the reference).
#define S2 4096
#define S1 4096
#define NN 1024

typedef __attribute__((ext_vector_type(2))) float v2f;
typedef __attribute__((ext_vector_type(8))) float v8f;

// ---------------------------------------------------------------------------
// NT GEMM with f32 WMMA (V_WMMA_F32_16X16X4_F32), wave32.
//   C[M, Nc] = A[M, K] * B[Nc, K]^T  (+ bias[Nc] if bias != nullptr)
// A and B both row-major, K contiguous -> per-lane fragments are contiguous
// float2's:
//   A frag: m = lane&15, k = kBase + 2*(lane>>4)  -> {A[m,k], A[m,k+1]}
//   B frag: n = lane&15, k = kBase + 2*(lane>>4)  -> {B[n,k], B[n,k+1]}
// C/D 16x16 f32 layout: lane gives N = lane&15; VGPR r gives M = 8*(lane>>4)+r.
//
// Each wave computes a 32x64 chunk (2 M-tiles x 4 N-tiles = 8 accumulators).
// The K loop processes K=16 per iteration: 24 *distinct* fragment registers
// are loaded up front so the scheduler can keep all global_load_b64's in
// flight and drain them with partial s_wait_loadcnt values instead of
// serializing each WMMA behind a single reused register pair.
// ---------------------------------------------------------------------------
__global__ void __launch_bounds__(128) gemm_nt_f32_wmma(
    const float* __restrict__ A, const float* __restrict__ B,
    const float* __restrict__ bias, float* __restrict__ C,
    int M, int Nc, int K) {
  const int lane   = threadIdx.x & 31;
  const int wave   = threadIdx.x >> 5;
  const int chunksX = Nc >> 6;                       // chunks are 64 wide
  const int chunk  = blockIdx.x * 4 + wave;
  const int mBase  = (chunk / chunksX) << 5;         // 32 rows per chunk
  const int nBase  = (chunk % chunksX) << 6;         // 64 cols per chunk
  if (mBase >= M) return;                            // wave-uniform guard

  const int r15 = lane & 15;
  const int kl  = (lane >> 4) << 1;                  // 0 or 2

  const float* aRow0 = A + (size_t)(mBase + r15) * (size_t)K + kl;
  const float* aRow1 = aRow0 + (size_t)16 * (size_t)K;
  const float* bRow0 = B + (size_t)(nBase +  0 + r15) * (size_t)K + kl;
  const float* bRow1 = B + (size_t)(nBase + 16 + r15) * (size_t)K + kl;
  const float* bRow2 = B + (size_t)(nBase + 32 + r15) * (size_t)K + kl;
  const float* bRow3 = B + (size_t)(nBase + 48 + r15) * (size_t)K + kl;

  v8f acc00 = {}, acc01 = {}, acc02 = {}, acc03 = {};
  v8f acc10 = {}, acc11 = {}, acc12 = {}, acc13 = {};

#pragma unroll 2
  for (int k = 0; k < K; k += 16) {
    // ---- load phase: 24 independent fragments (K=16 chunk) ----
    v2f a0f[4], a1f[4], bf0[4], bf1[4], bf2[4], bf3[4];
#pragma unroll
    for (int j = 0; j < 4; ++j) {
      const int kk = k + 4 * j;
      a0f[j] = *(const v2f*)(aRow0 + kk);
      a1f[j] = *(const v2f*)(aRow1 + kk);
      bf0[j] = *(const v2f*)(bRow0 + kk);
      bf1[j] = *(const v2f*)(bRow1 + kk);
      bf2[j] = *(const v2f*)(bRow2 + kk);
      bf3[j] = *(const v2f*)(bRow3 + kk);
    }
    // ---- math phase: 32 WMMAs over 8 independent accumulator chains ----
#pragma unroll
    for (int j = 0; j < 4; ++j) {
      // 8 args: (neg_a, A, neg_b, B, c_mod, C, reuse_a, reuse_b)
      acc00 = __builtin_amdgcn_wmma_f32_16x16x4_f32(false, a0f[j], false, bf0[j], (short)0, acc00, false, false);
      acc10 = __builtin_amdgcn_wmma_f32_16x16x4_f32(false, a1f[j], false, bf0[j], (short)0, acc10, false, false);
      acc01 = __builtin_amdgcn_wmma_f32_16x16x4_f32(false, a0f[j], false, bf1[j], (short)0, acc01, false, false);
      acc11 = __builtin_amdgcn_wmma_f32_16x16x4_f32(false, a1f[j], false, bf1[j], (short)0, acc11, false, false);
      acc02 = __builtin_amdgcn_wmma_f32_16x16x4_f32(false, a0f[j], false, bf2[j], (short)0, acc02, false, false);
      acc12 = __builtin_amdgcn_wmma_f32_16x16x4_f32(false, a1f[j], false, bf2[j], (short)0, acc12, false, false);
      acc03 = __builtin_amdgcn_wmma_f32_16x16x4_f32(false, a0f[j], false, bf3[j], (short)0, acc03, false, false);
      acc13 = __builtin_amdgcn_wmma_f32_16x16x4_f32(false, a1f[j], false, bf3[j], (short)0, acc13, false, false);
    }
  }

  // Store: row = mBase + ms*16 + 8*(lane>>4) + r, col = nBase + t*16 + (lane&15)
  const int rBase = (lane >> 4) << 3;                // 0 or 8
  v8f* accs[2][4] = {{&acc00, &acc01, &acc02, &acc03},
                     {&acc10, &acc11, &acc12, &acc13}};
#pragma unroll
  for (int ms = 0; ms < 2; ++ms) {
#pragma unroll
    for (int t = 0; t < 4; ++t) {
      const int col = nBase + t * 16 + r15;
      const float bv = bias ? bias[col] : 0.0f;
      float* cp = C + (size_t)(mBase + ms * 16 + rBase) * (size_t)Nc + col;
      v8f acc = *accs[ms][t];
#pragma unroll
      for (int r = 0; r < 8; ++r) {
        *cp = acc[r] + bv;
        cp += Nc;
      }
    }
  }
}

// ---------------------------------------------------------------------------
// Row softmax over a [rows, 4096] f32 matrix, in place. One block per row,
// 256 threads, 16 elements per thread held in registers; LDS tree reductions
// for max and sum (matches jax.nn.softmax: exp(x - max) / sum).
// ---------------------------------------------------------------------------
__global__ void __launch_bounds__(256) softmax_rows_4096(float* __restrict__ E) {
  const int tid = threadIdx.x;
  float* p = E + (size_t)blockIdx.x * (size_t)S1;

  float v[16];
  float m = -INFINITY;
#pragma unroll
  for (int i = 0; i < 16; ++i) {
    v[i] = p[tid + (i << 8)];
    m = fmaxf(m, v[i]);
  }

  __shared__ float red[256];
  red[tid] = m;
  __syncthreads();
#pragma unroll
  for (int s = 128; s > 0; s >>= 1) {
    if (tid < s) red[tid] = fmaxf(red[tid], red[tid + s]);
    __syncthreads();
  }
  m = red[0];
  __syncthreads();

  float sum = 0.0f;
#pragma unroll
  for (int i = 0; i < 16; ++i) {
    v[i] = __expf(v[i] - m);
    sum += v[i];
  }
  red[tid] = sum;
  __syncthreads();
#pragma unroll
  for (int s = 128; s > 0; s >>= 1) {
    if (tid < s) red[tid] += red[tid + s];
    __syncthreads();
  }
  const float inv = 1.0f / red[0];
#pragma unroll
  for (int i = 0; i < 16; ++i) {
    p[tid + (i << 8)] = v[i] * inv;
  }
}

// ---------------------------------------------------------------------------
// Launcher. d_in order: out_state[S2,N], history[S1,N], W[N,N], b[N].
// d_ws holds proj[S1,N] f32 (16 MB). Energies go straight into d_out and are
// softmax-normalized in place.
// ---------------------------------------------------------------------------
extern "C" void kernel_launch(void* const* d_in, const int* in_sizes, int n_in,
                              void* d_out, int out_size, void* d_ws, size_t ws_size,
                              hipStream_t stream) {
  (void)in_sizes; (void)n_in; (void)out_size; (void)ws_size;
  const float* out_state = (const float*)d_in[0];
  const float* history   = (const float*)d_in[1];
  const float* W         = (const float*)d_in[2];
  const float* b         = (const float*)d_in[3];
  float* proj = (float*)d_ws;          // S1 * NN f32 = 16 MB scratch
  float* E    = (float*)d_out;         // S2 * S1 f32

  // proj = history @ W^T + b : M=4096, Nc=1024, K=1024
  //   chunks = (4096/32)*(1024/64) = 2048 waves -> 512 blocks of 4 waves.
  gemm_nt_f32_wmma<<<512, 128, 0, stream>>>(history, W, b, proj,
                                            S1, NN, NN);

  // E = out_state @ proj^T : M=4096, Nc=4096, K=1024
  //   chunks = (4096/32)*(4096/64) = 8192 waves -> 2048 blocks.
  gemm_nt_f32_wmma<<<2048, 128, 0, stream>>>(out_state, proj, nullptr, E,
                                             S2, S1, NN);

  // Row softmax in place on d_out.
  softmax_rows_4096<<<S2, 256, 0, stream>>>(E);
}